// GAT_50414326121242
// MI455X (gfx1250) — compile-verified
//
#include <hip/hip_runtime.h>
#include <hip/hip_bf16.h>
#include <math.h>

#define N_NODES  10000
#define N_EDGES  320000
#define HIDDEN   256
#define HEADS    8
#define HEAD_DIM 32
#define NEG_SLOPE 0.2f

typedef __attribute__((ext_vector_type(2))) float v2f;
typedef __attribute__((ext_vector_type(8))) float v8f;

// ---------------------------------------------------------------------------
// Float atomic max via signed/unsigned int ordering trick (works for any sign
// mix given init = -FLT_MAX).
// ---------------------------------------------------------------------------
__device__ __forceinline__ void atomicMaxF32(float* addr, float v) {
    if (v >= 0.0f)
        atomicMax((int*)addr, __float_as_int(v));
    else
        atomicMin((unsigned int*)addr, __float_as_uint(v));
}

// ---------------------------------------------------------------------------
// C[M,N] = A[M,K] * B[K,N], M=10000, N=K=256, fp32 WMMA 16x16x4.
// One block (512 threads = 16 wave32) per 16-row strip of C; wave w computes
// the 16x16 tile at column block w. All 16 waves consume IDENTICAL A
// fragments, so A is staged once into LDS, pre-swizzled into per-lane WMMA
// fragment layout:
//   aFrag[i][l] = A[strip + (l&15)][4*i + 2*(l>>4) + {0,1}]   (v2f, 16 KB)
// Per-iteration read is one conflict-free ds_load_b64 (32 lanes x 8B = 64
// banks exactly). B stays in global: 256 KB total, L2-resident across blocks.
//
// WMMA frags (ISA 7.12.2):
//   A 16x4: lane<16 -> row=lane,   K={kk,kk+1}; lane>=16 -> row=lane-16, K={kk+2,kk+3}
//   B 4x16: VGPR0 = rows {kk,kk+2} over lane halves; VGPR1 = rows {kk+1,kk+3}
//   D 16x16: VGPR i -> M = i + 8*(lane>>4), N = lane&15
// ---------------------------------------------------------------------------
__global__ __launch_bounds__(512) void gat_gemm_wmma(const float* __restrict__ A,
                                                     const float* __restrict__ B,
                                                     float* __restrict__ C) {
    __shared__ v2f aFrag[64][32];                 // [k-step][lane], 16 KB

    const int lane  = threadIdx.x & 31;
    const int wave  = threadIdx.x >> 5;           // 0..15  == tileN
    const int tileM = blockIdx.x;                 // 0..624

    // Cooperative swizzled fill: entry (i, l) <- A[row l&15][4i + 2*(l>>4)..+1]
    for (int idx = threadIdx.x; idx < 64 * 32; idx += 512) {
        int i = idx >> 5;
        int l = idx & 31;
        const float* src = A + (size_t)(tileM * 16 + (l & 15)) * HIDDEN
                             + 4 * i + 2 * (l >> 4);
        aFrag[i][l] = *(const v2f*)src;
    }
    __syncthreads();

    const int koff = 2 * (lane >> 4);
    const int colB = wave * 16 + (lane & 15);

    v8f acc = {};
    for (int i = 0; i < 64; ++i) {
        const int kk = 4 * i;
        v2f a = aFrag[i][lane];
        v2f b;
        b.x = B[(size_t)(kk + koff)     * HIDDEN + colB];
        b.y = B[(size_t)(kk + koff + 1) * HIDDEN + colB];
        // speculative near-scope prefetch of the B column 4 K-steps ahead
        __builtin_prefetch(&B[(size_t)(kk + koff + 16) * HIDDEN + colB], 0, 3);
        acc = __builtin_amdgcn_wmma_f32_16x16x4_f32(
            /*neg_a=*/false, a, /*neg_b=*/false, b,
            /*c_mod=*/(short)0, acc, /*reuse_a=*/false, /*reuse_b=*/false);
    }

    const int half = lane >> 4;
    const int n    = wave * 16 + (lane & 15);
#pragma unroll
    for (int i = 0; i < 8; ++i) {
        int m = tileM * 16 + i + 8 * half;
        C[(size_t)m * HIDDEN + n] = acc[i];
    }
}

// ---------------------------------------------------------------------------
// Per (node, head): alpha_src/alpha_dst dot products + init softmax stats and
// zero the aggregation accumulator.
// ---------------------------------------------------------------------------
__global__ void gat_prep(const float* __restrict__ h,
                         const float* __restrict__ a_src,
                         const float* __restrict__ a_dst,
                         float* __restrict__ as, float* __restrict__ ad,
                         float* __restrict__ m,  float* __restrict__ denom,
                         float* __restrict__ acc) {
    int t = blockIdx.x * blockDim.x + threadIdx.x;
    if (t >= N_NODES * HEADS) return;
    int hh = t & (HEADS - 1);
    int n  = t >> 3;
    const float* hrow = h + (size_t)n * HIDDEN + hh * HEAD_DIM;
    const float* av   = a_src + hh * HEAD_DIM;
    const float* dv   = a_dst + hh * HEAD_DIM;
    float* ap = acc + (size_t)n * HIDDEN + hh * HEAD_DIM;
    float s0 = 0.0f, s1 = 0.0f;
#pragma unroll
    for (int c = 0; c < HEAD_DIM; ++c) {
        float v = hrow[c];
        s0 += v * av[c];
        s1 += v * dv[c];
        ap[c] = 0.0f;
    }
    as[t] = s0;
    ad[t] = s1;
    m[t]     = -3.402823466e38f;   // -FLT_MAX
    denom[t] = 0.0f;
}

// ---------------------------------------------------------------------------
// Edge pass 1: atomic segment-max of e = leakyrelu(as[src]+ad[dst]) over dst.
// One thread per (edge, head). e is NOT stored; it is recomputed in pass 2
// from the tiny L2-resident as/ad arrays (saves a 10 MB store + load).
// ---------------------------------------------------------------------------
__global__ void gat_edge1(const int* __restrict__ edges,
                          const float* __restrict__ as,
                          const float* __restrict__ ad,
                          float* __restrict__ m) {
    int t = blockIdx.x * blockDim.x + threadIdx.x;
    if (t >= N_EDGES * HEADS) return;
    int e  = t >> 3;
    int hh = t & (HEADS - 1);
    int s = edges[e];
    int d = edges[N_EDGES + e];
    float v = as[s * HEADS + hh] + ad[d * HEADS + hh];
    v = (v > 0.0f) ? v : NEG_SLOPE * v;
    atomicMaxF32(&m[d * HEADS + hh], v);
}

// ---------------------------------------------------------------------------
// Edge pass 2: ex = exp(e - m[dst]); denom[dst] += ex;
//              acc[dst,h,:] += ex * h[src,h,:]  (unnormalized message sum).
// One wave (32 lanes) per edge; lane == channel; loop over 8 heads.
// ---------------------------------------------------------------------------
__global__ __launch_bounds__(256) void gat_edge2(const int* __restrict__ edges,
                                                 const float* __restrict__ as,
                                                 const float* __restrict__ ad,
                                                 const float* __restrict__ m,
                                                 const float* __restrict__ h,
                                                 float* __restrict__ denom,
                                                 float* __restrict__ acc) {
    int wave = (blockIdx.x * blockDim.x + threadIdx.x) >> 5;
    int lane = threadIdx.x & 31;
    if (wave >= N_EDGES) return;
    int s = edges[wave];
    int d = edges[N_EDGES + wave];
    const float* hs = h   + (size_t)s * HIDDEN;
    float*       ao = acc + (size_t)d * HIDDEN;
#pragma unroll
    for (int hh = 0; hh < HEADS; ++hh) {
        float v = as[s * HEADS + hh] + ad[d * HEADS + hh];
        v = (v > 0.0f) ? v : NEG_SLOPE * v;
        float ex = expf(v - m[d * HEADS + hh]);
        if (lane == 0) atomicAdd(&denom[d * HEADS + hh], ex);
        atomicAdd(&ao[hh * HEAD_DIM + lane], ex * hs[hh * HEAD_DIM + lane]);
    }
}

// ---------------------------------------------------------------------------
// Finalize: out = acc / (denom + 1e-16) + bias; optional ELU. One thread per
// (node, channel). Safe in place (element-wise).
// ---------------------------------------------------------------------------
__global__ void gat_fin(const float* __restrict__ acc,
                        const float* __restrict__ denom,
                        const float* __restrict__ b,
                        float* __restrict__ out, int do_elu) {
    int t = blockIdx.x * blockDim.x + threadIdx.x;
    if (t >= N_NODES * HIDDEN) return;
    int n  = t >> 8;            // / HIDDEN
    int c  = t & (HIDDEN - 1);
    int hh = c >> 5;            // / HEAD_DIM
    float v = acc[t] / (denom[n * HEADS + hh] + 1e-16f) + b[c];
    if (do_elu) v = (v > 0.0f) ? v : expm1f(v);
    out[t] = v;
}

// ---------------------------------------------------------------------------
extern "C" void kernel_launch(void* const* d_in, const int* in_sizes, int n_in,
                              void* d_out, int out_size, void* d_ws, size_t ws_size,
                              hipStream_t stream) {
    const float* x     = (const float*)d_in[0];
    const int*   edges = (const int*)  d_in[1];
    const float* W1    = (const float*)d_in[2];
    const float* asv1  = (const float*)d_in[3];
    const float* adv1  = (const float*)d_in[4];
    const float* b1    = (const float*)d_in[5];
    const float* W2    = (const float*)d_in[6];
    const float* asv2  = (const float*)d_in[7];
    const float* adv2  = (const float*)d_in[8];
    const float* b2    = (const float*)d_in[9];
    float* out = (float*)d_out;

    char* p = (char*)d_ws;
    float* h    = (float*)p; p += sizeof(float) * (size_t)N_NODES * HIDDEN;  // h1, then h2
    float* z    = (float*)p; p += sizeof(float) * (size_t)N_NODES * HIDDEN;  // elu(layer1)
    float* acc  = (float*)p; p += sizeof(float) * (size_t)N_NODES * HIDDEN;  // layer1 agg
    float* as   = (float*)p; p += sizeof(float) * (size_t)N_NODES * HEADS;
    float* ad   = (float*)p; p += sizeof(float) * (size_t)N_NODES * HEADS;
    float* mx   = (float*)p; p += sizeof(float) * (size_t)N_NODES * HEADS;
    float* dn   = (float*)p; p += sizeof(float) * (size_t)N_NODES * HEADS;

    const dim3 blk(256);
    const int gemm_blocks = N_NODES / 16;                              // 625 x 512 thr
    const int prep_blocks = (N_NODES * HEADS + 255) / 256;
    const int e1_blocks   = (N_EDGES * HEADS + 255) / 256;
    const int e2_blocks   = (N_EDGES * 32 + 255) / 256;
    const int fin_blocks  = (N_NODES * HIDDEN + 255) / 256;

    // ---- layer 1 ----
    gat_gemm_wmma<<<gemm_blocks, 512, 0, stream>>>(x, W1, h);
    gat_prep<<<prep_blocks, blk, 0, stream>>>(h, asv1, adv1, as, ad, mx, dn, acc);
    gat_edge1<<<e1_blocks, blk, 0, stream>>>(edges, as, ad, mx);
    gat_edge2<<<e2_blocks, blk, 0, stream>>>(edges, as, ad, mx, h, dn, acc);
    gat_fin<<<fin_blocks, blk, 0, stream>>>(acc, dn, b1, z, /*elu=*/1);

    // ---- layer 2 (accumulate directly into d_out) ----
    gat_gemm_wmma<<<gemm_blocks, 512, 0, stream>>>(z, W2, h);
    gat_prep<<<prep_blocks, blk, 0, stream>>>(h, asv2, adv2, as, ad, mx, dn, out);
    gat_edge1<<<e1_blocks, blk, 0, stream>>>(edges, as, ad, mx);
    gat_edge2<<<e2_blocks, blk, 0, stream>>>(edges, as, ad, mx, h, dn, out);
    gat_fin<<<fin_blocks, blk, 0, stream>>>(out, dn, b2, out, /*elu=*/0);
}